// GCN_84559316124278
// MI455X (gfx1250) — compile-verified
//
#include <hip/hip_runtime.h>
#include <stdint.h>

// ---------- WMMA fragment types (fp32 16x16x4) ----------
typedef __attribute__((ext_vector_type(2))) float v2f;
typedef __attribute__((ext_vector_type(8))) float v8f;

// ---------------------------------------------------------------------------
// Generic fill
// ---------------------------------------------------------------------------
__global__ void fill_f32(float* __restrict__ p, float v, long long n) {
    long long i = (long long)blockIdx.x * blockDim.x + threadIdx.x;
    long long stride = (long long)gridDim.x * blockDim.x;
    for (; i < n; i += stride) p[i] = v;
}

// ---------------------------------------------------------------------------
// Degree count (deg pre-initialized to 1.0 for the self-loop)
// ---------------------------------------------------------------------------
__global__ void count_deg(const long long* __restrict__ dst, float* __restrict__ deg, int E) {
    int e = blockIdx.x * blockDim.x + threadIdx.x;
    if (e < E) atomicAdd(&deg[dst[e]], 1.0f);
}

__global__ void rsqrt_inplace(float* __restrict__ deg, int N) {
    int i = blockIdx.x * blockDim.x + threadIdx.x;
    if (i < N) deg[i] = rsqrtf(deg[i]);   // deg >= 1 always (self loop)
}

// ---------------------------------------------------------------------------
// GEMM: Out[M,128] = A[M,K] @ W[K,128] via V_WMMA_F32_16X16X4_F32
// Block: 256 threads = 8 waves. Block owns 16 rows; wave w owns col tile w.
// A tile staged in LDS with padded stride (K+1) to avoid bank conflicts.
// Fragment layouts per CDNA5 ISA 7.12.2:
//   A 16x4 : lane m=L%16, VGPR{0,1} hold k = 2*(L/16)+{0,1}
//   B 4x16 : lane n=L%16, VGPR{0,1} hold k = 2*(L/16)+{0,1}
//   D 16x16: lane n=L%16, VGPR v holds m = v + 8*(L/16)
// ---------------------------------------------------------------------------
template <int K>
__global__ __launch_bounds__(256) void gemm_n128_wmma(
    const float* __restrict__ A, const float* __restrict__ W,
    float* __restrict__ Out, int M)
{
    __shared__ float As[16 * (K + 1)];
    const int tid  = threadIdx.x;
    const int row0 = blockIdx.x * 16;

    // Stage 16xK A tile into LDS (guarded for generality; M%16==0 here)
    for (int i = tid; i < 16 * K; i += 256) {
        int r = i / K, c = i % K;
        float v = 0.0f;
        if (row0 + r < M) v = A[(size_t)(row0 + r) * K + c];
        As[r * (K + 1) + c] = v;
    }
    __syncthreads();

    const int wave = tid >> 5;          // 0..7 -> 16-col tile
    const int lane = tid & 31;
    const int col0 = wave * 16;
    const int half = lane >> 4;         // 0 or 1
    const int lm   = lane & 15;         // m for A-frag, n for B-frag/D

    v8f acc = {};
#pragma unroll
    for (int k0 = 0; k0 < K; k0 += 4) {
        const int ka = k0 + 2 * half;
        v2f a, b;
        a.x = As[lm * (K + 1) + ka];
        a.y = As[lm * (K + 1) + ka + 1];
        b.x = W[(size_t)ka * 128 + col0 + lm];
        b.y = W[(size_t)(ka + 1) * 128 + col0 + lm];
        acc = __builtin_amdgcn_wmma_f32_16x16x4_f32(
            /*neg_a=*/false, a, /*neg_b=*/false, b,
            /*c_mod=*/(short)0, acc, /*reuse_a=*/false, /*reuse_b=*/false);
    }

#pragma unroll
    for (int v = 0; v < 8; ++v) {
        int r = row0 + v + 8 * half;
        if (r < M) Out[(size_t)r * 128 + col0 + lm] = acc[v];
    }
}

// ---------------------------------------------------------------------------
// Edge scatter: one wave per edge; float4 gather of 128-ch source row,
// scaled by dinv[src]*dinv[dst], atomic-add into the destination row.
// ---------------------------------------------------------------------------
__global__ __launch_bounds__(256) void scatter_edges(
    const long long* __restrict__ src, const long long* __restrict__ dst,
    const float* __restrict__ dinv,
    const float* __restrict__ Hin, float* __restrict__ Hout, int E)
{
    int e = blockIdx.x * 8 + (threadIdx.x >> 5);
    if (e >= E) return;
    int lane = threadIdx.x & 31;
    long long s = src[e], d = dst[e];
    float nrm = dinv[s] * dinv[d];
    float4 v = reinterpret_cast<const float4*>(Hin + (size_t)s * 128)[lane];
    float* out = Hout + (size_t)d * 128 + lane * 4;
    atomicAdd(out + 0, v.x * nrm);
    atomicAdd(out + 1, v.y * nrm);
    atomicAdd(out + 2, v.z * nrm);
    atomicAdd(out + 3, v.w * nrm);
}

// ---------------------------------------------------------------------------
// Epilogue: Hagg = relu(Hagg + dinv[n]^2 * Hin + bias). One wave per node.
// (folds the self-loop contribution so it never touches atomics)
// ---------------------------------------------------------------------------
__global__ __launch_bounds__(256) void self_bias_relu(
    float* __restrict__ Hagg, const float* __restrict__ Hin,
    const float* __restrict__ dinv, const float* __restrict__ bias, int N)
{
    int n = blockIdx.x * 8 + (threadIdx.x >> 5);
    if (n >= N) return;
    int lane = threadIdx.x & 31;
    int c = lane * 4;
    float w = dinv[n] * dinv[n];
    float4 a = *reinterpret_cast<float4*>(Hagg + (size_t)n * 128 + c);
    float4 h = *reinterpret_cast<const float4*>(Hin + (size_t)n * 128 + c);
    float4 r;
    r.x = fmaxf(a.x + w * h.x + bias[c + 0], 0.0f);
    r.y = fmaxf(a.y + w * h.y + bias[c + 1], 0.0f);
    r.z = fmaxf(a.z + w * h.z + bias[c + 2], 0.0f);
    r.w = fmaxf(a.w + w * h.w + bias[c + 3], 0.0f);
    *reinterpret_cast<float4*>(Hagg + (size_t)n * 128 + c) = r;
}

// ---------------------------------------------------------------------------
// Mean-pool accumulation: one wave per node -> pooled[batch[n]] and counts.
// ---------------------------------------------------------------------------
__global__ __launch_bounds__(256) void pool_accum(
    const float* __restrict__ H, const long long* __restrict__ batch,
    float* __restrict__ pooled, float* __restrict__ cnt, int N)
{
    int n = blockIdx.x * 8 + (threadIdx.x >> 5);
    if (n >= N) return;
    int lane = threadIdx.x & 31;
    long long g = batch[n];
    if (lane == 0) atomicAdd(&cnt[g], 1.0f);
    int c = lane * 4;
    float4 v = *reinterpret_cast<const float4*>(H + (size_t)n * 128 + c);
    float* p = pooled + (size_t)g * 128 + c;
    atomicAdd(p + 0, v.x);
    atomicAdd(p + 1, v.y);
    atomicAdd(p + 2, v.z);
    atomicAdd(p + 3, v.w);
}

// ---------------------------------------------------------------------------
// Classifier: out[64,10] = (pooled/cnt) @ Wc + bc   (tiny; scalar is fine)
// ---------------------------------------------------------------------------
__global__ void classify(const float* __restrict__ pooled, const float* __restrict__ cnt,
                         const float* __restrict__ Wc, const float* __restrict__ bc,
                         float* __restrict__ out, int nGraphs, int nClasses)
{
    int i = blockIdx.x * blockDim.x + threadIdx.x;
    if (i >= nGraphs * nClasses) return;
    int g = i / nClasses, c = i % nClasses;
    float inv = 1.0f / fmaxf(cnt[g], 1.0f);
    float s = bc[c];
#pragma unroll 4
    for (int k = 0; k < 128; ++k)
        s += pooled[(size_t)g * 128 + k] * inv * Wc[(size_t)k * nClasses + c];
    out[i] = s;
}

// ---------------------------------------------------------------------------
extern "C" void kernel_launch(void* const* d_in, const int* in_sizes, int n_in,
                              void* d_out, int out_size, void* d_ws, size_t ws_size,
                              hipStream_t stream)
{
    const float*     x    = (const float*)d_in[0];      // [N,64]
    const float*     W1   = (const float*)d_in[1];      // [64,128]
    const float*     b1   = (const float*)d_in[2];      // [128]
    const float*     W2   = (const float*)d_in[3];      // [128,128]
    const float*     b2   = (const float*)d_in[4];      // [128]
    const float*     Wc   = (const float*)d_in[5];      // [128,10]
    const float*     bc   = (const float*)d_in[6];      // [10]
    const long long* edge = (const long long*)d_in[7];  // [2,E] int64
    const long long* bat  = (const long long*)d_in[8];  // [N]   int64

    const int N = in_sizes[8];
    const int E = in_sizes[7] / 2;
    const int HID = 128;
    const int NG = 64, NC = 10;

    const long long* srcIdx = edge;       // edge_index[0]
    const long long* dstIdx = edge + E;   // edge_index[1]

    // ---- workspace layout (256B aligned) ----
    char* ws = (char*)d_ws;
    size_t off = 0;
    auto alloc = [&](size_t bytes) {
        char* p = ws + off;
        off = (off + bytes + 255) & ~(size_t)255;
        return p;
    };
    float* H0     = (float*)alloc((size_t)N * HID * sizeof(float)); // transform out
    float* H1     = (float*)alloc((size_t)N * HID * sizeof(float)); // aggregate out
    float* dinv   = (float*)alloc((size_t)N * sizeof(float));
    float* pooled = (float*)alloc((size_t)NG * HID * sizeof(float));
    float* cnt    = (float*)alloc((size_t)NG * sizeof(float));
    (void)ws_size;

    const int rowTiles   = (N + 15) / 16;            // 6250
    const int waveBlocks = (N + 7) / 8;              // node-per-wave kernels
    const int edgeBlocks = (E + 7) / 8;              // edge-per-wave kernels

    // ---- normalization coefficients ----
    fill_f32<<<2048, 256, 0, stream>>>(dinv, 1.0f, (long long)N);   // self loop
    count_deg<<<(E + 255) / 256, 256, 0, stream>>>(dstIdx, dinv, E);
    rsqrt_inplace<<<(N + 255) / 256, 256, 0, stream>>>(dinv, N);

    // ---- layer 1: H0 = x @ W1 ; H1 = scatter(H0) ; relu(+self+b1) ----
    gemm_n128_wmma<64><<<rowTiles, 256, 0, stream>>>(x, W1, H0, N);
    fill_f32<<<4096, 256, 0, stream>>>(H1, 0.0f, (long long)N * HID);
    scatter_edges<<<edgeBlocks, 256, 0, stream>>>(srcIdx, dstIdx, dinv, H0, H1, E);
    self_bias_relu<<<waveBlocks, 256, 0, stream>>>(H1, H0, dinv, b1, N);

    // ---- layer 2: H0 = H1 @ W2 ; H1 = scatter(H0) ; relu(+self+b2) ----
    gemm_n128_wmma<128><<<rowTiles, 256, 0, stream>>>(H1, W2, H0, N);
    fill_f32<<<4096, 256, 0, stream>>>(H1, 0.0f, (long long)N * HID);
    scatter_edges<<<edgeBlocks, 256, 0, stream>>>(srcIdx, dstIdx, dinv, H0, H1, E);
    self_bias_relu<<<waveBlocks, 256, 0, stream>>>(H1, H0, dinv, b2, N);

    // ---- global mean pool + classifier ----
    fill_f32<<<64, 256, 0, stream>>>(pooled, 0.0f, (long long)NG * HID);
    fill_f32<<<1, 64, 0, stream>>>(cnt, 0.0f, (long long)NG);
    pool_accum<<<waveBlocks, 256, 0, stream>>>(H1, bat, pooled, cnt, N);
    classify<<<(NG * NC + 255) / 256, 256, 0, stream>>>(pooled, cnt, Wc, bc,
                                                        (float*)d_out, NG, NC);
}